// HistoryOwnGainModulator_72112500900254
// MI455X (gfx1250) — compile-verified
//
#include <hip/hip_runtime.h>

#define B_ 512
#define N_ 8192
#define H_ 5
#define S_ 10
#define T_ 5000

typedef __attribute__((ext_vector_type(2))) float v2f;
typedef __attribute__((ext_vector_type(8))) float v8f;

// ---- batch gain: windowed gather-dot of own_gain at sort_id, then elu()+1 ----
// window w[k]=0 for k<100, w[100+j]=exp(-0.5*(j/30)^2); idx = s+100-k = s-j.
__global__ __launch_bounds__(256) void hog_gain_kernel(const float* __restrict__ own_gain,
                                                       const int* __restrict__ sort_id,
                                                       float* __restrict__ batch_gain) {
  int b = blockIdx.x * blockDim.x + threadIdx.x;
  if (b >= B_) return;
  int s = sort_id[b];
  float sum = 0.f;
#pragma unroll 4
  for (int j = 0; j <= 100; ++j) {
    int idx = s - j;
    float g = (idx >= 0 && idx < T_) ? own_gain[idx] : 0.f;
    float t = (float)j * (1.0f / 30.0f);
    float w = __expf(-0.5f * t * t);
    sum = fmaf(g, w, sum);
  }
  float e = (sum > 0.f) ? sum : (__expf(sum) - 1.f);
  batch_gain[b] = e + 1.f;
}

// ---- fused main kernel ----
// Each wave: one 16(b) x 16(n) tile.
//   state_mod tile = WMMA_F32_16x16x4_F32 over K=10 (padded to 12), C seeded with state_b[n].
//   Then per lane 8 output elements: x + hist_dot + elu(state_mod); elu()+1; gain scale.
__global__ __launch_bounds__(128) void hog_main_kernel(
    const float* __restrict__ x,
    const float* __restrict__ history,
    const float* __restrict__ state,
    const float* __restrict__ history_weights,
    const float* __restrict__ history_bias,
    const float* __restrict__ state_W,
    const float* __restrict__ state_b,
    const float* __restrict__ gain_adjust,
    const float* __restrict__ batch_gain,
    float* __restrict__ out) {
  const int lane = threadIdx.x & 31;
  const int wave = threadIdx.x >> 5;
  const int nloc = lane & 15;
  const int hi   = lane >> 4;              // lane half: 0 (lanes 0-15) / 1 (lanes 16-31)
  const int n0 = blockIdx.x * 64 + wave * 16;
  const int b0 = blockIdx.y * 16;
  const int n  = n0 + nloc;

  // per-n constants (same for all 8 rows this lane touches)
  const float w0 = history_weights[n * H_ + 0];
  const float w1 = history_weights[n * H_ + 1];
  const float w2 = history_weights[n * H_ + 2];
  const float w3 = history_weights[n * H_ + 3];
  const float w4 = history_weights[n * H_ + 4];
  const float bias_n = history_bias[n];
  const float ga_n   = gain_adjust[n];
  const float sb_n   = state_b[n];

  // ---- state GEMM tile via f32 WMMA ----
  // C/D layout: vgpr i, lanes 0-15 -> (M=i, N=lane); lanes 16-31 -> (M=8+i, N=lane-16)
  v8f acc;
#pragma unroll
  for (int i = 0; i < 8; ++i) acc[i] = sb_n;

  const int m = nloc;  // A-matrix row (batch within tile) = lane%16
#pragma unroll
  for (int c = 0; c < 3; ++c) {
    const int k0 = 4 * c + 2 * hi;       // A/B K-pair base per lane half
    // Branch-free K padding: clamp the index (loads always in-bounds),
    // then zero the value with a select. Avoids exec-mask branches between WMMAs.
    const int ka = (k0     < S_) ? k0     : 0;
    const int kb = (k0 + 1 < S_) ? k0 + 1 : 0;
    const bool va = (k0     < S_);
    const bool vb = (k0 + 1 < S_);
    v2f a, bf;
    float ax = state[(b0 + m) * S_ + ka];
    float ay = state[(b0 + m) * S_ + kb];
    float bx = state_W[n * S_ + ka];      // B[k][n] = state_W[n][k]
    float by = state_W[n * S_ + kb];
    a.x  = va ? ax : 0.f;
    a.y  = vb ? ay : 0.f;
    bf.x = va ? bx : 0.f;
    bf.y = vb ? by : 0.f;
    acc = __builtin_amdgcn_wmma_f32_16x16x4_f32(
        /*neg_a=*/false, a, /*neg_b=*/false, bf,
        /*c_mod=*/(short)0, acc, /*reuse_a=*/false, /*reuse_b=*/false);
  }

  // ---- fused elementwise pass over the 16x16 tile (8 elements per lane) ----
#pragma unroll
  for (int i = 0; i < 8; ++i) {
    const int b = b0 + hi * 8 + i;
    const size_t base = (size_t)b * N_ + n;
    const float xv = x[base];
    const float* hp = history + base * H_;
    float hv = bias_n;
    hv = fmaf(hp[0], w0, hv);
    hv = fmaf(hp[1], w1, hv);
    hv = fmaf(hp[2], w2, hv);
    hv = fmaf(hp[3], w3, hv);
    hv = fmaf(hp[4], w4, hv);
    const float sm  = acc[i];
    const float esm = (sm > 0.f) ? sm : (__expf(sm) - 1.f);   // elu(state_mod)
    const float t   = xv + hv + esm;
    const float et  = (t > 0.f) ? t : (__expf(t) - 1.f);      // elu(x)
    const float xv2 = et + 1.f;
    const float bg  = batch_gain[b];                           // already elu()+1
    out[base] = fmaf(ga_n * bg, xv2, xv2);                     // (1 + ga*bg) * xv2
  }
}

extern "C" void kernel_launch(void* const* d_in, const int* in_sizes, int n_in,
                              void* d_out, int out_size, void* d_ws, size_t ws_size,
                              hipStream_t stream) {
  const float* x               = (const float*)d_in[0];
  const float* history         = (const float*)d_in[1];
  // d_in[2] = gain (unused by reference)
  const float* state           = (const float*)d_in[3];
  const int*   sort_id         = (const int*)d_in[4];
  const float* history_weights = (const float*)d_in[5];
  const float* history_bias    = (const float*)d_in[6];
  const float* state_W         = (const float*)d_in[7];
  const float* state_b         = (const float*)d_in[8];
  const float* own_gain        = (const float*)d_in[9];
  const float* gain_adjust     = (const float*)d_in[10];
  float* out = (float*)d_out;
  float* batch_gain = (float*)d_ws;   // 512 floats of scratch

  hipLaunchKernelGGL(hog_gain_kernel, dim3((B_ + 255) / 256), dim3(256), 0, stream,
                     own_gain, sort_id, batch_gain);
  dim3 grid(N_ / 64, B_ / 16);
  hipLaunchKernelGGL(hog_main_kernel, grid, dim3(128), 0, stream,
                     x, history, state, history_weights, history_bias,
                     state_W, state_b, gain_adjust, batch_gain, out);
}